// GATWithFeatureAttention_9045201125951
// MI455X (gfx1250) — compile-verified
//
#include <hip/hip_runtime.h>

// ---------------- problem constants (from the reference) ----------------
#define N_NODES   50000
#define N_EDGES   800000
#define IN_CH     128
#define HIDC      64
#define OUT_CH    32
#define HEADS     4
#define H1DIM     (HEADS * HIDC)      // 256
#define NEG_SLOPE 0.2f
#define EPSV      1e-16f

typedef float v2f __attribute__((ext_vector_type(2)));
typedef float v8f __attribute__((ext_vector_type(8)));

// ---------------- WMMA wrapper: native fp32 tensor op on CDNA5 ----------------
__device__ inline v8f wmma_f32_4(v2f a, v2f b, v8f c) {
#if __has_builtin(__builtin_amdgcn_wmma_f32_16x16x4_f32)
  // (neg_a, A, neg_b, B, c_mod, C, reuse_a, reuse_b) -> v_wmma_f32_16x16x4_f32
  return __builtin_amdgcn_wmma_f32_16x16x4_f32(false, a, false, b, (short)0, c,
                                               false, false);
#else
  // compile-safe placeholder (would show wmma=0 in histogram -> iterate)
  for (int i = 0; i < 8; ++i) c[i] += a.x * b.x + a.y * b.y;
  return c;
#endif
}

// ---------------- small helpers ----------------
__device__ inline float leakyf(float x) { return x > 0.f ? x : NEG_SLOPE * x; }
__device__ inline float eluf(float x)   { return x > 0.f ? x : __expf(x) - 1.f; }

// ordered-bits float atomic max: lowers to global_atomic_max_i32 / min_u32
__device__ inline void atomicMaxF(float* addr, float v) {
  if (v >= 0.f) atomicMax((int*)addr, __float_as_int(v));
  else          atomicMin((unsigned int*)addr, (unsigned int)__float_as_int(v));
}

__global__ void fill_f32(float* __restrict__ p, float v, int n) {
  int t = blockIdx.x * blockDim.x + threadIdx.x;
  if (t < n) p[t] = v;
}
__global__ void copy_i32(const int* __restrict__ s, int* __restrict__ d, int n) {
  int t = blockIdx.x * blockDim.x + threadIdx.x;
  if (t < n) d[t] = s[t];
}

// ---------------- GEMM1: h1raw[N,256] = x[N,128] @ W1[128,256] ----------------
// One block = 16 rows. 4 waves; each wave owns 4 column tiles (16 cols each).
// A tile staged in LDS (8KB) and reused by all 16 wmma streams in the block.
__global__ __launch_bounds__(128) void gemm1_wmma(const float* __restrict__ A,
                                                  const float* __restrict__ B,
                                                  float* __restrict__ C) {
  __shared__ float As[16 * IN_CH];
  const int tid  = threadIdx.x;
  const int wave = tid >> 5;
  const int lane = tid & 31;
  const int half = lane >> 4;   // which half-wave (K split)
  const int m    = lane & 15;
  const int row0 = blockIdx.x * 16;

  for (int i = tid; i < 16 * IN_CH; i += 128) {
    int r = i >> 7, c = i & (IN_CH - 1);
    As[i] = A[(size_t)(row0 + r) * IN_CH + c];
  }
  __syncthreads();

  v8f acc[4];
#pragma unroll
  for (int j = 0; j < 4; ++j)
#pragma unroll
    for (int i = 0; i < 8; ++i) acc[j][i] = 0.f;

  for (int k0 = 0; k0 < IN_CH; k0 += 4) {
    v2f a;                                       // A 16x4 frag (ISA layout)
    a.x = As[m * IN_CH + k0 + 2 * half + 0];
    a.y = As[m * IN_CH + k0 + 2 * half + 1];
#pragma unroll
    for (int j = 0; j < 4; ++j) {
      const int col = wave * 16 + j * 64;
      v2f b;                                     // B 4x16 frag
      b.x = B[(size_t)(k0 + 2 * half + 0) * H1DIM + col + m];
      b.y = B[(size_t)(k0 + 2 * half + 1) * H1DIM + col + m];
      acc[j] = wmma_f32_4(a, b, acc[j]);
    }
  }
#pragma unroll
  for (int j = 0; j < 4; ++j) {
    const int col = wave * 16 + j * 64 + m;
#pragma unroll
    for (int i = 0; i < 8; ++i)
      C[(size_t)(row0 + i + half * 8) * H1DIM + col] = acc[j][i];
  }
}

// ---------------- GEMM2: h2raw[N,32] = h1act[N,256] @ W2[256,32] ----------------
__global__ __launch_bounds__(64) void gemm2_wmma(const float* __restrict__ A,
                                                 const float* __restrict__ B,
                                                 float* __restrict__ C) {
  __shared__ float As[16 * H1DIM];  // 16KB
  const int tid  = threadIdx.x;
  const int wave = tid >> 5;
  const int lane = tid & 31;
  const int half = lane >> 4;
  const int m    = lane & 15;
  const int row0 = blockIdx.x * 16;

  for (int i = tid; i < 16 * H1DIM; i += 64) {
    int r = i >> 8, c = i & (H1DIM - 1);
    As[i] = A[(size_t)(row0 + r) * H1DIM + c];
  }
  __syncthreads();

  v8f acc;
#pragma unroll
  for (int i = 0; i < 8; ++i) acc[i] = 0.f;
  const int col = wave * 16;

  for (int k0 = 0; k0 < H1DIM; k0 += 4) {
    v2f a, b;
    a.x = As[m * H1DIM + k0 + 2 * half + 0];
    a.y = As[m * H1DIM + k0 + 2 * half + 1];
    b.x = B[(size_t)(k0 + 2 * half + 0) * OUT_CH + col + m];
    b.y = B[(size_t)(k0 + 2 * half + 1) * OUT_CH + col + m];
    acc = wmma_f32_4(a, b, acc);
  }
#pragma unroll
  for (int i = 0; i < 8; ++i)
    C[(size_t)(row0 + i + half * 8) * OUT_CH + col + m] = acc[i];
}

// ---------------- attention coefficients, layer 1 ----------------
// block = node, wave = head; 32 lanes x float2 = 64 channels; shfl reduce.
__global__ __launch_bounds__(128) void att1_kernel(
    const float* __restrict__ h1raw, const float* __restrict__ att_src,
    const float* __restrict__ att_dst, float* __restrict__ a_src,
    float* __restrict__ a_dst, float* __restrict__ m1, float* __restrict__ s1) {
  const int n = blockIdx.x;
  const int h = threadIdx.x >> 5;
  const int lane = threadIdx.x & 31;
  const float2 v  = *(const float2*)(h1raw + (size_t)n * H1DIM + h * HIDC + lane * 2);
  const float2 as = *(const float2*)(att_src + h * HIDC + lane * 2);
  const float2 ad = *(const float2*)(att_dst + h * HIDC + lane * 2);
  float ss = v.x * as.x + v.y * as.y;
  float dd = v.x * ad.x + v.y * ad.y;
  for (int off = 16; off > 0; off >>= 1) {
    ss += __shfl_xor(ss, off, 32);
    dd += __shfl_xor(dd, off, 32);
  }
  if (lane == 0) {
    a_src[n * HEADS + h] = ss;
    a_dst[n * HEADS + h] = dd;
    m1[n * HEADS + h] = -__builtin_inff();
    s1[n * HEADS + h] = 0.f;
  }
}

// ---------------- attention coefficients, layer 2 (1 head, 32ch) ----------------
__global__ __launch_bounds__(128) void att2_kernel(
    const float* __restrict__ h2raw, const float* __restrict__ att_src,
    const float* __restrict__ att_dst, float* __restrict__ a_src,
    float* __restrict__ a_dst, float* __restrict__ m2, float* __restrict__ s2) {
  const int w = (blockIdx.x * blockDim.x + threadIdx.x) >> 5;
  const int lane = threadIdx.x & 31;
  if (w >= N_NODES) return;
  const float v = h2raw[(size_t)w * OUT_CH + lane];
  float ss = v * att_src[lane];
  float dd = v * att_dst[lane];
  for (int off = 16; off > 0; off >>= 1) {
    ss += __shfl_xor(ss, off, 32);
    dd += __shfl_xor(dd, off, 32);
  }
  if (lane == 0) {
    a_src[w] = ss; a_dst[w] = dd;
    m2[w] = -__builtin_inff(); s2[w] = 0.f;
  }
}

// ---------------- edge passes, layer 1 ----------------
__global__ void edge1_max(const int* __restrict__ src, const int* __restrict__ dst,
                          const float* __restrict__ a_src, const float* __restrict__ a_dst,
                          float* __restrict__ m1) {
  int t = blockIdx.x * blockDim.x + threadIdx.x;
  if (t >= N_EDGES * HEADS) return;
  int e = t >> 2, h = t & 3;
  __builtin_prefetch(&src[e + 2048], 0, 0);
  int s = src[e], d = dst[e];
  float ev = leakyf(a_src[s * HEADS + h] + a_dst[d * HEADS + h]);
  atomicMaxF(&m1[d * HEADS + h], ev);
}

__global__ void edge1_sum(const int* __restrict__ src, const int* __restrict__ dst,
                          const float* __restrict__ a_src, const float* __restrict__ a_dst,
                          const float* __restrict__ m1, float* __restrict__ s1) {
  int t = blockIdx.x * blockDim.x + threadIdx.x;
  if (t >= N_EDGES * HEADS) return;
  int e = t >> 2, h = t & 3;
  __builtin_prefetch(&src[e + 2048], 0, 0);
  int s = src[e], d = dst[e];
  float ev = leakyf(a_src[s * HEADS + h] + a_dst[d * HEADS + h]);
  atomicAdd(&s1[d * HEADS + h], __expf(ev - m1[d * HEADS + h]));
}

// wave = (edge, head): 32 lanes x float2 = 64 channels scatter-added.
__global__ __launch_bounds__(256) void edge1_agg(
    const int* __restrict__ src, const int* __restrict__ dst,
    const float* __restrict__ a_src, const float* __restrict__ a_dst,
    const float* __restrict__ m1, const float* __restrict__ s1,
    const float* __restrict__ h1raw, float* __restrict__ h1acc,
    float* __restrict__ alpha_out) {
  const int w = (blockIdx.x * 256 + threadIdx.x) >> 5;
  const int lane = threadIdx.x & 31;
  const int e = w >> 2, h = w & 3;
  if (e >= N_EDGES) return;
  const int s = src[e], d = dst[e];
  const float ev = leakyf(a_src[s * HEADS + h] + a_dst[d * HEADS + h]);
  const float ex = __expf(ev - m1[d * HEADS + h]);
  const float alpha = ex / (s1[d * HEADS + h] + EPSV);
  if (lane == 0) alpha_out[(size_t)e * HEADS + h] = alpha;
  const float2 hv = *(const float2*)(h1raw + (size_t)s * H1DIM + h * HIDC + lane * 2);
  float* op = h1acc + (size_t)d * H1DIM + h * HIDC + lane * 2;
  atomicAdd(op + 0, hv.x * alpha);
  atomicAdd(op + 1, hv.y * alpha);
}

// ---------------- bias + ELU (layer1 output -> layer2 input) ----------------
__global__ void finalize1(float* __restrict__ h1acc, const float* __restrict__ bias1) {
  int t = blockIdx.x * blockDim.x + threadIdx.x;
  if (t >= N_NODES * H1DIM) return;
  h1acc[t] = eluf(h1acc[t] + bias1[t & (H1DIM - 1)]);
}

// ---------------- edge passes, layer 2 ----------------
__global__ void edge2_max(const int* __restrict__ src, const int* __restrict__ dst,
                          const float* __restrict__ a_src, const float* __restrict__ a_dst,
                          float* __restrict__ m2) {
  int e = blockIdx.x * blockDim.x + threadIdx.x;
  if (e >= N_EDGES) return;
  __builtin_prefetch(&src[e + 2048], 0, 0);
  float ev = leakyf(a_src[src[e]] + a_dst[dst[e]]);
  atomicMaxF(&m2[dst[e]], ev);
}

__global__ void edge2_sum(const int* __restrict__ src, const int* __restrict__ dst,
                          const float* __restrict__ a_src, const float* __restrict__ a_dst,
                          const float* __restrict__ m2, float* __restrict__ s2) {
  int e = blockIdx.x * blockDim.x + threadIdx.x;
  if (e >= N_EDGES) return;
  int d = dst[e];
  float ev = leakyf(a_src[src[e]] + a_dst[d]);
  atomicAdd(&s2[d], __expf(ev - m2[d]));
}

// wave = edge: 32 lanes = 32 channels, accumulate straight into d_out h2 region.
__global__ __launch_bounds__(256) void edge2_agg(
    const int* __restrict__ src, const int* __restrict__ dst,
    const float* __restrict__ a_src, const float* __restrict__ a_dst,
    const float* __restrict__ m2, const float* __restrict__ s2,
    const float* __restrict__ h2raw, float* __restrict__ out_h2,
    float* __restrict__ alpha_out) {
  const int e = (blockIdx.x * 256 + threadIdx.x) >> 5;
  const int lane = threadIdx.x & 31;
  if (e >= N_EDGES) return;
  const int s = src[e], d = dst[e];
  const float ev = leakyf(a_src[s] + a_dst[d]);
  const float ex = __expf(ev - m2[d]);
  const float alpha = ex / (s2[d] + EPSV);
  if (lane == 0) alpha_out[e] = alpha;
  atomicAdd(&out_h2[(size_t)d * OUT_CH + lane],
            h2raw[(size_t)s * OUT_CH + lane] * alpha);
}

__global__ void finalize2(float* __restrict__ out_h2, const float* __restrict__ bias2) {
  int t = blockIdx.x * blockDim.x + threadIdx.x;
  if (t >= N_NODES * OUT_CH) return;
  out_h2[t] += bias2[t & (OUT_CH - 1)];
}

// ---------------- host orchestration ----------------
extern "C" void kernel_launch(void* const* d_in, const int* in_sizes, int n_in,
                              void* d_out, int out_size, void* d_ws, size_t ws_size,
                              hipStream_t stream) {
  const float* x        = (const float*)d_in[0];
  const int*   ei       = (const int*)  d_in[1];
  const float* W1       = (const float*)d_in[2];
  const float* att_src1 = (const float*)d_in[3];
  const float* att_dst1 = (const float*)d_in[4];
  const float* bias1    = (const float*)d_in[5];
  const float* W2       = (const float*)d_in[6];
  const float* att_src2 = (const float*)d_in[7];
  const float* att_dst2 = (const float*)d_in[8];
  const float* bias2    = (const float*)d_in[9];
  (void)in_sizes; (void)n_in; (void)out_size; (void)ws_size;

  const int* src = ei;
  const int* dst = ei + N_EDGES;

  // workspace layout (floats)
  float* ws      = (float*)d_ws;
  float* h1raw   = ws;                                    // N*256
  float* h1acc   = h1raw + (size_t)N_NODES * H1DIM;       // N*256
  float* h2raw   = h1acc + (size_t)N_NODES * H1DIM;       // N*32
  float* a_src1v = h2raw + (size_t)N_NODES * OUT_CH;      // N*4
  float* a_dst1v = a_src1v + N_NODES * HEADS;             // N*4
  float* m1      = a_dst1v + N_NODES * HEADS;             // N*4
  float* s1      = m1 + N_NODES * HEADS;                  // N*4
  float* a_src2v = s1 + N_NODES * HEADS;                  // N
  float* a_dst2v = a_src2v + N_NODES;                     // N
  float* m2      = a_dst2v + N_NODES;                     // N
  float* s2      = m2 + N_NODES;                          // N

  // output layout: h2 | edge_index | alpha1 | edge_index | alpha2
  float* out       = (float*)d_out;
  float* out_h2    = out;                                   // 1,600,000
  int*   out_ei1   = (int*)(out + 1600000);                 // 1,600,000
  float* out_alph1 = out + 3200000;                         // 3,200,000
  int*   out_ei2   = (int*)(out + 6400000);                 // 1,600,000
  float* out_alph2 = out + 8000000;                         //   800,000

  // ---- init / edge_index passthrough ----
  fill_f32<<<(N_NODES * H1DIM + 255) / 256, 256, 0, stream>>>(h1acc, 0.f, N_NODES * H1DIM);
  fill_f32<<<(N_NODES * OUT_CH + 255) / 256, 256, 0, stream>>>(out_h2, 0.f, N_NODES * OUT_CH);
  copy_i32<<<(2 * N_EDGES + 255) / 256, 256, 0, stream>>>(ei, out_ei1, 2 * N_EDGES);
  copy_i32<<<(2 * N_EDGES + 255) / 256, 256, 0, stream>>>(ei, out_ei2, 2 * N_EDGES);

  // ---- layer 1 ----
  gemm1_wmma<<<N_NODES / 16, 128, 0, stream>>>(x, W1, h1raw);
  att1_kernel<<<N_NODES, 128, 0, stream>>>(h1raw, att_src1, att_dst1,
                                           a_src1v, a_dst1v, m1, s1);
  edge1_max<<<(N_EDGES * HEADS + 255) / 256, 256, 0, stream>>>(src, dst, a_src1v, a_dst1v, m1);
  edge1_sum<<<(N_EDGES * HEADS + 255) / 256, 256, 0, stream>>>(src, dst, a_src1v, a_dst1v, m1, s1);
  edge1_agg<<<(N_EDGES * HEADS * 32 + 255) / 256, 256, 0, stream>>>(
      src, dst, a_src1v, a_dst1v, m1, s1, h1raw, h1acc, out_alph1);
  finalize1<<<(N_NODES * H1DIM + 255) / 256, 256, 0, stream>>>(h1acc, bias1);

  // ---- layer 2 ----
  gemm2_wmma<<<N_NODES / 16, 64, 0, stream>>>(h1acc, W2, h2raw);
  att2_kernel<<<(N_NODES * 32 + 127) / 128, 128, 0, stream>>>(h2raw, att_src2, att_dst2,
                                                              a_src2v, a_dst2v, m2, s2);
  edge2_max<<<(N_EDGES + 255) / 256, 256, 0, stream>>>(src, dst, a_src2v, a_dst2v, m2);
  edge2_sum<<<(N_EDGES + 255) / 256, 256, 0, stream>>>(src, dst, a_src2v, a_dst2v, m2, s2);
  edge2_agg<<<(N_EDGES * 32 + 255) / 256, 256, 0, stream>>>(
      src, dst, a_src2v, a_dst2v, m2, s2, h2raw, out_h2, out_alph2);
  finalize2<<<(N_NODES * OUT_CH + 255) / 256, 256, 0, stream>>>(out_h2, bias2);
}